// ShapeletBlocks_81389630259716
// MI455X (gfx1250) — compile-verified
//
#include <hip/hip_runtime.h>
#include <math.h>

typedef __attribute__((ext_vector_type(2))) float v2f;
typedef __attribute__((ext_vector_type(8))) float v8f;
typedef __attribute__((ext_vector_type(4))) int   v4i;

#define AS1 __attribute__((address_space(1)))
#define AS3 __attribute__((address_space(3)))

constexpr int BB   = 64;    // batch
constexpr int CH   = 32;    // channels
constexpr int TT   = 1024;  // time
constexpr int NSHP = 256;   // shapelets per size
constexpr int FOUT = 768;   // 3 * NSHP
constexpr int NCHK = 64;    // n-columns per workgroup
constexpr int WSTR = 72;    // W-tile LDS stride: 2*72 % 64 == 16 -> half-waves on disjoint banks
constexpr int FSTR = 68;    // f-tile LDS stride: 8*68 % 64 == 32 -> halves disjoint

#if __has_builtin(__builtin_amdgcn_global_load_async_to_lds_b128)
#define USE_ASYNC 1
#else
#define USE_ASYNC 0
#endif

__device__ __forceinline__ void wait_async0() {
#if USE_ASYNC
#if __has_builtin(__builtin_amdgcn_s_wait_asynccnt)
  __builtin_amdgcn_s_wait_asynccnt(0);
#else
  asm volatile("s_wait_asynccnt 0" ::: "memory");
#endif
#endif
}

// Stage x window buf[c][j] = x[b][c][t0 - S + j], j in [0, S+32).
// Async path: 16B granules; OOB (t<0) granules are skipped entirely (S,t0,j all
// multiples of 4), their slots are pre-zeroed once before the loop.
template <int S>
__device__ __forceinline__ void stage_window(const float* __restrict__ xb,
                                             float* buf, int t0, int tid) {
  constexpr int XR = S + 32;
#if USE_ASYNC
  for (int i = 4 * tid; i < CH * XR; i += 4 * 128) {
    int c = i / XR, j = i % XR;
    int t = t0 - S + j;
    if (t >= 0) {
      __builtin_amdgcn_global_load_async_to_lds_b128(
          (AS1 v4i*)(xb + c * TT + t), (AS3 v4i*)(buf + i),
          /*offset=*/0, /*cpol=*/0);
    }
  }
#else
  for (int i = tid; i < CH * XR; i += 128) {
    int c = i / XR, j = i % XR;
    int t = t0 - S + j;
    buf[i] = (t >= 0) ? xb[c * TT + t] : 0.0f;
  }
#endif
}

// out[b][t][fbase+n] = max_{t'<=t} ( bias[n] + sum_{c,k} x[b][c][t'+k-S]*W[n][c][k] )
template <int S>
__global__ __launch_bounds__(128) void shapelet_wmma(
    const float* __restrict__ x, const float* __restrict__ W,
    const float* __restrict__ bias, float* __restrict__ out, int fbase) {
  constexpr int K  = CH * S;   // GEMM depth
  constexpr int XR = S + 32;   // window row length (covers 32 t' rows)

  extern __shared__ float smem[];
  float* wl  = smem;              // [K][WSTR]  staged weights, K-major
  float* xw0 = wl + K * WSTR;     // [CH][XR]   window buffer 0
  float* xw1 = xw0 + CH * XR;     // [CH][XR]   window buffer 1
  float* ft  = xw1 + CH * XR;     // [32][FSTR] f tile for cummax handoff

  const int b    = blockIdx.x;
  const int n0   = blockIdx.y * NCHK;
  const int tid  = threadIdx.x;
  const int lane = tid & 31;
  const int wv   = tid >> 5;      // wave 0..3 -> n sub-strip of 16
  const int hf   = lane >> 4;     // half-wave (f32 fragment layout, ISA 7.12.2)
  const int ln   = lane & 15;
  const int nw   = wv * 16;

  // ---- stage W slice: wl[k][nn] = W[n0+nn][c][kin], k = c*S+kin (contiguous
  //      global reads as float4, scatter into K-major LDS) ----
  const float* Ws = W + (size_t)n0 * K;
  for (int i = 4 * tid; i < NCHK * K; i += 4 * 128) {
    const float4 v = *(const float4*)(Ws + i);
    int nn = i / K, k = i % K;
    wl[(k + 0) * WSTR + nn] = v.x;
    wl[(k + 1) * WSTR + nn] = v.y;
    wl[(k + 2) * WSTR + nn] = v.z;
    wl[(k + 3) * WSTR + nn] = v.w;
  }

  // ---- pre-zero the left-pad slots of buffer 0 (only tile 0 has t<0; the
  //      async loader never writes these slots) ----
  for (int i = tid; i < CH * S; i += 128) {
    int c = i / S, j = i % S;
    xw0[c * XR + j] = 0.0f;
  }

  float bias_r = 0.0f;
  float carry  = -INFINITY;       // running cummax per n across all tiles
  if (tid < NCHK) bias_r = bias[n0 + tid];

  const float* xb = x + (size_t)b * CH * TT;
  float* ob = out + (size_t)b * TT * FOUT + fbase + n0;

  // prologue: kick off window for tile 0
  stage_window<S>(xb, xw0, 0, tid);

  for (int it = 0; it < TT / 32; ++it) {
    const int t0 = it * 32;
    wait_async0();        // our wave's async writes to cur are done
    __syncthreads();      // ...and everyone's; prev iter's LDS readers drained

    float* cur = (it & 1) ? xw1 : xw0;
    float* nxt = (it & 1) ? xw0 : xw1;
    if (it + 1 < TT / 32) stage_window<S>(xb, nxt, t0 + 32, tid);  // prefetch

    // ---- implicit GEMM: two 16x16 M-tiles per wave share each B fragment ----
    v8f acc0 = {0.f, 0.f, 0.f, 0.f, 0.f, 0.f, 0.f, 0.f};
    v8f acc1 = {0.f, 0.f, 0.f, 0.f, 0.f, 0.f, 0.f, 0.f};
    const float* xbase = cur + ln + 2 * hf;              // A: lane M=ln, K=k0+2h
    const float* wbase = wl + 2 * hf * WSTR + nw + ln;   // B: lane N=ln, K=k0+2h
    for (int c = 0; c < CH; ++c) {
      const float* xr = xbase + c * XR;
      const float* wr = wbase + (c * S) * WSTR;
#pragma unroll
      for (int kin = 0; kin < S; kin += 4) {
        v2f a0, a1, bf;
        a0.x = xr[kin];      a0.y = xr[kin + 1];       // rows t0..t0+15
        a1.x = xr[kin + 16]; a1.y = xr[kin + 17];      // rows t0+16..t0+31
        bf.x = wr[kin * WSTR];
        bf.y = wr[(kin + 1) * WSTR];
        acc0 = __builtin_amdgcn_wmma_f32_16x16x4_f32(false, a0, false, bf,
                                                     (short)0, acc0, false, false);
        acc1 = __builtin_amdgcn_wmma_f32_16x16x4_f32(false, a1, false, bf,
                                                     (short)0, acc1, false, false);
      }
    }

    // ---- spill D tiles: VGPR j -> row j + 8*hf (+16 for second tile) ----
#pragma unroll
    for (int j = 0; j < 8; ++j) {
      ft[(j + 8 * hf) * FSTR + nw + ln]      = acc0[j];
      ft[(16 + j + 8 * hf) * FSTR + nw + ln] = acc1[j];
    }
    __syncthreads();

    // ---- fused cummax + streaming store: one thread per n column ----
    if (tid < NCHK) {
      float* op = ob + (size_t)t0 * FOUT + tid;
#pragma unroll
      for (int tt = 0; tt < 32; ++tt) {
        carry = fmaxf(carry, ft[tt * FSTR + tid] + bias_r);
        __builtin_nontemporal_store(carry, op + (size_t)tt * FOUT);
      }
    }
  }
}

static inline size_t smem_bytes(int S) {
  return (size_t)((32 * S) * WSTR + 2 * 32 * (S + 32) + 32 * FSTR) * sizeof(float);
}

extern "C" void kernel_launch(void* const* d_in, const int* in_sizes, int n_in,
                              void* d_out, int out_size, void* d_ws, size_t ws_size,
                              hipStream_t stream) {
  const float* x   = (const float*)d_in[0];
  const float* W8  = (const float*)d_in[1];
  const float* b8  = (const float*)d_in[2];
  const float* W16 = (const float*)d_in[3];
  const float* b16 = (const float*)d_in[4];
  const float* W32 = (const float*)d_in[5];
  const float* b32 = (const float*)d_in[6];
  float* out = (float*)d_out;

  dim3 grid(BB, NSHP / NCHK);  // (64, 4) per size
  shapelet_wmma<8 ><<<grid, 128, smem_bytes(8),  stream>>>(x, W8,  b8,  out, 0);
  shapelet_wmma<16><<<grid, 128, smem_bytes(16), stream>>>(x, W16, b16, out, 256);
  shapelet_wmma<32><<<grid, 128, smem_bytes(32), stream>>>(x, W32, b32, out, 512);
}